// MF_28432683500220
// MI455X (gfx1250) — compile-verified
//
#include <hip/hip_runtime.h>
#include <hip/hip_bf16.h>

typedef __attribute__((ext_vector_type(2))) float v2f;
typedef __attribute__((ext_vector_type(8))) float v8f;

#define MF_BATCH 16384
#define MF_EMB   128

// One wave32 computes 16 (user,item) dot products via V_WMMA_F32_16X16X4_F32.
// A = 16 user rows (16xK), B = 16 item rows transposed (Kx16); diag(A.B) are
// the wanted row-wise dots, accumulated in fp32 across K=128 in 32 WMMAs.
__global__ __launch_bounds__(256) void MF_28432683500220_kernel(
    const float* __restrict__ user_emb,
    const float* __restrict__ user_bias,
    const float* __restrict__ item_emb,
    const float* __restrict__ item_bias,
    const float* __restrict__ mean,
    const int*   __restrict__ u_id,
    const int*   __restrict__ i_id,
    float*       __restrict__ out)
{
    const int lane     = threadIdx.x & 31;
    const int wave     = blockIdx.x * (blockDim.x >> 5) + (threadIdx.x >> 5);
    const int tileBase = wave * 16;                  // 16 pairs per wave
    if (tileBase >= MF_BATCH) return;                // wave-uniform; EXEC stays all-ones

    const int row   = lane & 15;                     // M (== N) served by this lane
    const int khalf = (lane >> 4) << 1;              // hi half-wave carries K+2,K+3

    const int u  = u_id[tileBase + row];
    const int it = i_id[tileBase + row];

    const float* __restrict__ Urow = user_emb + (size_t)u  * MF_EMB;
    const float* __restrict__ Irow = item_emb + (size_t)it * MF_EMB;

    v8f acc = {};
    #pragma unroll
    for (int k0 = 0; k0 < MF_EMB; k0 += 4) {
        v2f a = *(const v2f*)(Urow + k0 + khalf);    // A: 16x4 fp32 tile
        v2f b = *(const v2f*)(Irow + k0 + khalf);    // B: 4x16 fp32 tile (item rows as columns)
        // 8 args: (neg_a, A, neg_b, B, c_mod, C, reuse_a, reuse_b)
        acc = __builtin_amdgcn_wmma_f32_16x16x4_f32(
            false, a, false, b, (short)0, acc, false, false);
    }

    // Diagonal of the 16x16 fp32 C/D tile:
    //   n in 0..7  -> VGPR n,   lane n        (lanes 0..7)
    //   n in 8..15 -> VGPR n-8, lane n+16     (lanes 24..31)
    const bool active = (lane < 8) | (lane >= 24);
    if (active) {
        const int r = lane & 7;
        float d = acc[r];
        const int n = (lane < 8) ? lane : (lane - 16);
        const int bidx = tileBase + n;               // == tileBase + row for these lanes
        d += user_bias[u] + item_bias[it] + mean[0];
        out[bidx] = d;
    }
}

extern "C" void kernel_launch(void* const* d_in, const int* in_sizes, int n_in,
                              void* d_out, int out_size, void* d_ws, size_t ws_size,
                              hipStream_t stream) {
    const float* user_emb  = (const float*)d_in[0];
    const float* user_bias = (const float*)d_in[1];
    const float* item_emb  = (const float*)d_in[2];
    const float* item_bias = (const float*)d_in[3];
    const float* mean      = (const float*)d_in[4];
    const int*   u_id      = (const int*)d_in[5];
    const int*   i_id      = (const int*)d_in[6];
    float*       out       = (float*)d_out;

    // 16 pairs per wave, 8 waves per 256-thread block -> 128 pairs per block.
    const int pairsPerBlock = (256 / 32) * 16;
    const int blocks = (MF_BATCH + pairsPerBlock - 1) / pairsPerBlock;  // = 128
    MF_28432683500220_kernel<<<blocks, 256, 0, stream>>>(
        user_emb, user_bias, item_emb, item_bias, mean, u_id, i_id, out);
}